// SAGE_15075335209145
// MI455X (gfx1250) — compile-verified
//
#include <hip/hip_runtime.h>
#include <hip/hip_bf16.h>
#include <math.h>

typedef __attribute__((ext_vector_type(2))) float v2f;
typedef __attribute__((ext_vector_type(8))) float v8f;

#define DIN 128  // feature width of every aggregation input

// ---------------------------------------------------------------------------
// Degree: one thread per edge, f32 atomic histogram on dst.
// ---------------------------------------------------------------------------
__global__ void sage_degree(const int* __restrict__ dst, float* __restrict__ deg, int E) {
    int e = blockIdx.x * blockDim.x + threadIdx.x;
    if (e < E) atomicAdd(&deg[dst[e]], 1.0f);
}

__global__ void sage_invdeg(float* __restrict__ deg, int n) {
    int i = blockIdx.x * blockDim.x + threadIdx.x;
    if (i < n) deg[i] = 1.0f / fmaxf(deg[i], 1.0f);
}

// ---------------------------------------------------------------------------
// Edge scatter: one wave (32 lanes) per edge. Lane loads float4 of the source
// row (32*16B = 512B = full 128-float row) and atomically accumulates into the
// destination row. Node table (25.6MB) is L2-resident on MI455X (192MB L2),
// so both the gather and the f32 atomics resolve in L2.
// ---------------------------------------------------------------------------
__global__ void sage_scatter(const float* __restrict__ x,
                             const int* __restrict__ src,
                             const int* __restrict__ dst,
                             float* __restrict__ agg, int E) {
    int gid  = blockIdx.x * blockDim.x + threadIdx.x;
    int wid  = gid >> 5;
    int lane = gid & 31;
    if (wid >= E) return;
    int s = src[wid];
    int d = dst[wid];
    const float4* xs = (const float4*)(x + (size_t)s * DIN);
    float4 v = xs[lane];
    float* ad = agg + (size_t)d * DIN + lane * 4;
    atomicAdd(ad + 0, v.x);
    atomicAdd(ad + 1, v.y);
    atomicAdd(ad + 2, v.z);
    atomicAdd(ad + 3, v.w);
}

// ---------------------------------------------------------------------------
// Fused SAGE layer GEMM: out[16 x MOUT tile] = (agg*invdeg)@Wl + self@Wr + b.
// Block = (MOUT/16) waves; wave w computes the 16x16 tile at columns 16w.
// Combined A ([agg | self], 16 x 256) staged in LDS with pitch 17 floats:
//  - staging writes (consecutive kk, fixed r) stride 17 dwords -> conflict-free
//  - WMMA A-fragment reads (lanes 0-15: r=0..15 at fixed kk) consecutive,
//    lanes 16-31 offset by 34 dwords -> disjoint banks.
// K loop: 64 x v_wmma_f32_16x16x4_f32 per wave (fp32-exact path; GEMM flops
// are only ~8 GFLOP total so fp32 matrix rate is more than sufficient).
// ---------------------------------------------------------------------------
template <int MOUT, bool RELU>
__global__ void sage_gemm(const float* __restrict__ agg,
                          const float* __restrict__ selfin,
                          const float* __restrict__ invdeg,
                          const float* __restrict__ Wl,
                          const float* __restrict__ Wr,
                          const float* __restrict__ bias,
                          float* __restrict__ out) {
    __shared__ float Alds[256 * 17];   // 17408 B

    const int row0 = blockIdx.x * 16;
    const int tid  = threadIdx.x;

    // Stage A = [agg-rows (scaled by invdeg) | self-rows], coalesced over kk.
    for (int idx = tid; idx < 16 * DIN * 2; idx += blockDim.x) {
        int part = idx >> 11;          // 0 = agg half, 1 = self half
        int rem  = idx & 2047;
        int r    = rem >> 7;           // row within tile
        int kk   = rem & 127;          // column
        float v;
        if (part == 0) {
            v = agg[(size_t)(row0 + r) * DIN + kk] * invdeg[row0 + r];
        } else {
            v = selfin[(size_t)(row0 + r) * DIN + kk];
        }
        Alds[(part * DIN + kk) * 17 + r] = v;
    }
    __syncthreads();

    const int lane = tid & 31;
    const int wave = tid >> 5;
    const int r    = lane & 15;        // row (A) / col-in-tile (B,D)
    const int h    = lane >> 4;        // K half-select within the 16x16x4 op
    const int col  = wave * 16 + r;    // global output column for this lane

    v8f acc = {};

    // Phase 1: (agg*invdeg) @ Wl   (K = 0..127 of combined A)
#pragma unroll
    for (int k = 0; k < 128; k += 4) {
        int kk = k + 2 * h;
        v2f a = { Alds[kk * 17 + r], Alds[(kk + 1) * 17 + r] };
        v2f b = { Wl[kk * MOUT + col], Wl[(kk + 1) * MOUT + col] };
        acc = __builtin_amdgcn_wmma_f32_16x16x4_f32(false, a, false, b,
                                                    (short)0, acc, false, false);
    }
    // Phase 2: self @ Wr           (K = 128..255 of combined A)
#pragma unroll
    for (int k = 0; k < 128; k += 4) {
        int kk = k + 2 * h;
        v2f a = { Alds[(128 + kk) * 17 + r], Alds[(128 + kk + 1) * 17 + r] };
        v2f b = { Wr[kk * MOUT + col], Wr[(kk + 1) * MOUT + col] };
        acc = __builtin_amdgcn_wmma_f32_16x16x4_f32(false, a, false, b,
                                                    (short)0, acc, false, false);
    }

    const float bv = bias[col];
#pragma unroll
    for (int i = 0; i < 8; ++i) {
        float v = acc[i] + bv;
        if (RELU) v = fmaxf(v, 0.0f);
        out[(size_t)(row0 + i + 8 * h) * MOUT + col] = v;   // D: VGPR i -> row i+8h
    }
}

// ---------------------------------------------------------------------------
// log_softmax over 64 columns: one wave per row, 2 elements per lane,
// shuffle-based max/sum reductions (wave32).
// ---------------------------------------------------------------------------
__global__ void sage_logsoftmax(float* __restrict__ out, int n) {
    int gid  = blockIdx.x * blockDim.x + threadIdx.x;
    int row  = gid >> 5;
    int lane = gid & 31;
    if (row >= n) return;
    float* p = out + (size_t)row * 64;
    float v0 = p[lane];
    float v1 = p[lane + 32];
    float m = fmaxf(v0, v1);
#pragma unroll
    for (int off = 16; off > 0; off >>= 1) m = fmaxf(m, __shfl_xor(m, off, 32));
    float s = __expf(v0 - m) + __expf(v1 - m);
#pragma unroll
    for (int off = 16; off > 0; off >>= 1) s += __shfl_xor(s, off, 32);
    float ls = m + __logf(s);
    p[lane]      = v0 - ls;
    p[lane + 32] = v1 - ls;
}

// ---------------------------------------------------------------------------
extern "C" void kernel_launch(void* const* d_in, const int* in_sizes, int n_in,
                              void* d_out, int out_size, void* d_ws, size_t ws_size,
                              hipStream_t stream) {
    const float* x   = (const float*)d_in[0];
    const int*   src = (const int*)d_in[1];
    const int*   dst = (const int*)d_in[2];
    const float* Wl0 = (const float*)d_in[3];
    const float* bl0 = (const float*)d_in[4];
    const float* Wr0 = (const float*)d_in[5];
    const float* Wl1 = (const float*)d_in[6];
    const float* bl1 = (const float*)d_in[7];
    const float* Wr1 = (const float*)d_in[8];
    const float* Wl2 = (const float*)d_in[9];
    const float* bl2 = (const float*)d_in[10];
    const float* Wr2 = (const float*)d_in[11];
    float* outF = (float*)d_out;

    const int N = in_sizes[0] / DIN;   // 50000
    const int E = in_sizes[1];         // 800000

    // Workspace: invdeg | agg | act  (~51.4 MB)
    char* ws = (char*)d_ws;
    size_t degBytes = ((size_t)N * 4 + 511) & ~(size_t)511;
    size_t aggBytes = (size_t)N * DIN * 4;
    float* deg = (float*)ws;
    float* agg = (float*)(ws + degBytes);
    float* act = (float*)(ws + degBytes + aggBytes);

    const int TPB = 256;

    // Degrees -> reciprocal (computed every call; deterministic).
    hipMemsetAsync(deg, 0, (size_t)N * 4, stream);
    sage_degree<<<(E + TPB - 1) / TPB, TPB, 0, stream>>>(dst, deg, E);
    sage_invdeg<<<(N + TPB - 1) / TPB, TPB, 0, stream>>>(deg, N);

    const int scatterBlocks = (E * 32 + TPB - 1) / TPB;   // one wave per edge
    const int rowTiles = N / 16;                          // 3125

    // Layer 0: agg(x) ; act = relu(agg@Wl0 + bl0 + x@Wr0)
    hipMemsetAsync(agg, 0, aggBytes, stream);
    sage_scatter<<<scatterBlocks, TPB, 0, stream>>>(x, src, dst, agg, E);
    sage_gemm<128, true><<<rowTiles, 256, 0, stream>>>(agg, x, deg, Wl0, Wr0, bl0, act);

    // Layer 1: agg(act) ; act = relu(agg@Wl1 + bl1 + act@Wr1)   (in-place safe)
    hipMemsetAsync(agg, 0, aggBytes, stream);
    sage_scatter<<<scatterBlocks, TPB, 0, stream>>>(act, src, dst, agg, E);
    sage_gemm<128, true><<<rowTiles, 256, 0, stream>>>(agg, act, deg, Wl1, Wr1, bl1, act);

    // Layer 2: agg(act) ; out = agg@Wl2 + bl2 + act@Wr2  (no relu), then log_softmax
    hipMemsetAsync(agg, 0, aggBytes, stream);
    sage_scatter<<<scatterBlocks, TPB, 0, stream>>>(act, src, dst, agg, E);
    sage_gemm<64, false><<<rowTiles, 128, 0, stream>>>(agg, act, deg, Wl2, Wr2, bl2, outF);

    sage_logsoftmax<<<(N * 32 + TPB - 1) / TPB, TPB, 0, stream>>>(outF, N);
}